// MVPLoss_38405597561574
// MI455X (gfx1250) — compile-verified
//
#include <hip/hip_runtime.h>
#include <hip/hip_bf16.h>
#include <math.h>

// Problem constants from the reference
#define N_PTS 128
#define DIMS  8192
#define MARGIN_F 200.0f

typedef __attribute__((ext_vector_type(2))) float v2f;
typedef __attribute__((ext_vector_type(8))) float v8f;

// ---------------------------------------------------------------------------
// Kernel 1: squared row norms of f1 (rows 0..127) and f2 (rows 128..255)
// ---------------------------------------------------------------------------
__global__ void row_norms_kernel(const float* __restrict__ f1,
                                 const float* __restrict__ f2,
                                 float* __restrict__ norms) {
    const int row = blockIdx.x;               // 0..255
    const float* src = (row < N_PTS) ? (f1 + (size_t)row * DIMS)
                                     : (f2 + (size_t)(row - N_PTS) * DIMS);
    float s = 0.0f;
    for (int i = threadIdx.x; i < DIMS; i += blockDim.x) {
        float v = src[i];
        s += v * v;
    }
    __shared__ float red[256];
    red[threadIdx.x] = s;
    __syncthreads();
    for (int off = 128; off > 0; off >>= 1) {
        if ((int)threadIdx.x < off) red[threadIdx.x] += red[threadIdx.x + off];
        __syncthreads();
    }
    if (threadIdx.x == 0) norms[row] = red[0];
}

// ---------------------------------------------------------------------------
// Kernel 2: GM tile via V_WMMA_F32_16X16X4_F32.
// One wave (32 lanes) per 16x16 output tile; 64 tiles total.
//
// 32-bit A-matrix 16x4 layout (ISA 7.12.2): lane L<16 holds row M=L with
// VGPR0=K0, VGPR1=K1; lane L>=16 holds row M=L-16 with VGPR0=K2, VGPR1=K3.
// B (4x16, = f2 tile transposed) mirrors this with lane = N, so both operands
// use the identical per-lane addressing into their row-major source rows.
// ---------------------------------------------------------------------------
__global__ void gm_wmma_kernel(const float* __restrict__ f1,
                               const float* __restrict__ f2,
                               const int* __restrict__ t1,
                               const int* __restrict__ t2,
                               const float* __restrict__ norms,
                               float* __restrict__ GM) {
    const int tm = blockIdx.x >> 3;          // tile row  (0..7)
    const int tn = blockIdx.x & 7;           // tile col  (0..7)
    const int lane = threadIdx.x;            // 0..31
    const int half = lane >> 4;              // 0 or 1
    const int lr   = lane & 15;              // row/col within tile

    const float* arow = f1 + (size_t)(tm * 16 + lr) * DIMS + 2 * half;
    const float* brow = f2 + (size_t)(tn * 16 + lr) * DIMS + 2 * half;

    v8f acc = {};
#pragma unroll 8
    for (int k = 0; k < DIMS; k += 4) {
        v2f a = *(const v2f*)(arow + k);
        v2f b = *(const v2f*)(brow + k);
        acc = __builtin_amdgcn_wmma_f32_16x16x4_f32(
            /*neg_a=*/false, a, /*neg_b=*/false, b,
            /*c_mod=*/(short)0, acc, /*reuse_a=*/false, /*reuse_b=*/false);
    }

    // C/D layout: VGPR r, lanes 0-15 -> M=r, lanes 16-31 -> M=8+r; N = lane%16
#pragma unroll
    for (int r = 0; r < 8; ++r) {
        const int m = tm * 16 + r + 8 * half;   // f1 row
        const int n = tn * 16 + lr;             // f2 row
        const float dot = acc[r];
        const float d2  = norms[m] - 2.0f * dot + norms[N_PTS + n];
        const bool  sim = (t1[m] == t2[n]);
        // SAME_MARGIN = 0: positive pairs -> max(0, d2); negative -> max(0, 200 - d2)
        const float gm = sim ? fmaxf(0.0f, d2) : fmaxf(0.0f, MARGIN_F - d2);
        GM[m * N_PTS + n] = gm;
    }
}

// ---------------------------------------------------------------------------
// Kernel 3: faithful single-thread port of the recursive Kuhn-Munkres
// (float64 labels/slacks, same 1e-5 tolerances, DFS via explicit stack),
// then loss = (1/n) * sum_y GM[link[y], y].
// ---------------------------------------------------------------------------
__global__ void km_loss_kernel(const float* __restrict__ GM,
                               float* __restrict__ out) {
    if (threadIdx.x != 0 || blockIdx.x != 0) return;

    const int n = N_PTS;
    double lx[N_PTS], ly[N_PTS], sla[N_PTS];
    int link[N_PTS], visx[N_PTS], visy[N_PTS];
    int stk_x[N_PTS + 1], stk_y[N_PTS + 1];

    for (int i = 0; i < n; ++i) {
        link[i] = -1;
        ly[i] = 0.0;
        double mx = (double)GM[i * n];
        for (int j = 1; j < n; ++j) {
            double v = (double)GM[i * n + j];
            if (v > mx) mx = v;
        }
        lx[i] = mx;
    }

    for (int x0 = 0; x0 < n; ++x0) {
        for (int j = 0; j < n; ++j) sla[j] = 1e10;
        for (;;) {
            for (int j = 0; j < n; ++j) { visx[j] = 0; visy[j] = 0; }

            // ---- iterative dfs(x0) ----
            int sp = 0;
            stk_x[0] = x0; stk_y[0] = 0;
            visx[x0] = 1;
            bool success = false;
            while (sp >= 0) {
                const int x = stk_x[sp];
                int y = stk_y[sp];
                int action = 0;              // 0=exhausted, 1=descended, 2=success
                for (; y < n; ++y) {
                    if (visy[y]) continue;
                    const double tmp = lx[x] + ly[y] - (double)GM[x * n + y];
                    if (fabs(tmp) < 1e-5) {
                        visy[y] = 1;
                        if (link[y] == -1) {
                            link[y] = x;
                            action = 2;
                            break;
                        } else {
                            stk_y[sp] = y;   // remember descend edge
                            const int nx = link[y];
                            ++sp;
                            stk_x[sp] = nx; stk_y[sp] = 0;
                            visx[nx] = 1;
                            action = 1;
                            break;
                        }
                    } else if (sla[y] + 1e-5 > tmp) {
                        sla[y] = tmp;
                    }
                }
                if (action == 2) {
                    while (sp > 0) {         // unwind: reassign along the path
                        --sp;
                        link[stk_y[sp]] = stk_x[sp];
                    }
                    success = true;
                    break;
                } else if (action == 1) {
                    continue;
                } else {
                    --sp;                    // frame failed
                    if (sp >= 0) stk_y[sp] += 1;  // resume parent past its edge
                }
            }
            if (success) break;

            // ---- dual update ----
            double d = 1e300;
            for (int j = 0; j < n; ++j)
                if (!visy[j] && sla[j] < d) d = sla[j];
            for (int j = 0; j < n; ++j) if (visx[j]) lx[j] -= d;
            for (int j = 0; j < n; ++j) {
                if (visy[j]) ly[j] += d;
                else         sla[j] -= d;
            }
        }
    }

    double loss = 0.0;
    for (int y = 0; y < n; ++y)
        if (link[y] != -1)
            loss += (double)GM[link[y] * n + y];
    out[0] = (float)(loss / (double)n);
}

// ---------------------------------------------------------------------------
extern "C" void kernel_launch(void* const* d_in, const int* in_sizes, int n_in,
                              void* d_out, int out_size, void* d_ws, size_t ws_size,
                              hipStream_t stream) {
    const float* f1 = (const float*)d_in[0];
    const float* f2 = (const float*)d_in[1];
    const int*   t1 = (const int*)d_in[2];
    const int*   t2 = (const int*)d_in[3];

    float* GM    = (float*)d_ws;                              // 128*128*4 = 64 KB
    float* norms = (float*)((char*)d_ws + N_PTS * N_PTS * 4); // 256*4 = 1 KB

    row_norms_kernel<<<2 * N_PTS, 256, 0, stream>>>(f1, f2, norms);
    gm_wmma_kernel<<<(N_PTS / 16) * (N_PTS / 16), 32, 0, stream>>>(
        f1, f2, t1, t2, norms, GM);
    km_loss_kernel<<<1, 1, 0, stream>>>(GM, (float*)d_out);
}